// GCN_11768210391434
// MI455X (gfx1250) — compile-verified
//
#include <hip/hip_runtime.h>
#include <math.h>

// MaxK-GNN forward for MI455X (gfx1250), wave32, fp32 WMMA 16x16x4.
//
// Layout assumptions (CDNA5 ISA 7.12.2):
//   A (16x4 f32): lane m in [0,16) holds A[m][k0+0..1], lane m+16 holds A[m][k0+2..3]
//   B (4x16 f32): symmetric -> lane n holds B[k0+2*hl + v][n] in VGPR v
//   C/D (16x16 f32): acc[v] = C[v + 8*hl][lane&15]

#define FEAT 128     // IN == HID == 128
#define OUTF 64
#define KSEL 32      // MaxK top-k
#define ASTR (FEAT + 4)   // LDS row stride (pad 4 floats -> conflict-free b64 reads)

typedef __attribute__((ext_vector_type(2))) float v2f;
typedef __attribute__((ext_vector_type(8))) float v8f;

// ---------------------------------------------------------------- utilities
__global__ void zero4_kernel(float4* __restrict__ p, int n4) {
    int i = blockIdx.x * blockDim.x + threadIdx.x;
    if (i < n4) p[i] = make_float4(0.f, 0.f, 0.f, 0.f);
}

__global__ void zero1_kernel(float* __restrict__ p, int n) {
    int i = blockIdx.x * blockDim.x + threadIdx.x;
    if (i < n) p[i] = 0.f;
}

// counts out-degree into dou[], in-degree into din[] (as floats)
__global__ void degree_count_kernel(const int* __restrict__ src, const int* __restrict__ dst,
                                    float* __restrict__ dou, float* __restrict__ din, int E) {
    int e = blockIdx.x * blockDim.x + threadIdx.x;
    if (e < E) {
        atomicAdd(&dou[src[e]], 1.f);
        atomicAdd(&din[dst[e]], 1.f);
    }
}

// in-place: d = 1/sqrt(max(d,1))
__global__ void degree_finalize_kernel(float* __restrict__ dou, float* __restrict__ din, int N) {
    int i = blockIdx.x * blockDim.x + threadIdx.x;
    if (i < N) {
        dou[i] = 1.f / sqrtf(fmaxf(dou[i], 1.f));
        din[i] = 1.f / sqrtf(fmaxf(din[i], 1.f));
    }
}

// h[i] = h[i] * din[row] + bg[col]     (vectorized float4; 32 float4 per row)
__global__ void spmm_finalize_kernel(float4* __restrict__ h4, const float* __restrict__ din,
                                     const float* __restrict__ bg, int n4) {
    int i = blockIdx.x * blockDim.x + threadIdx.x;
    if (i >= n4) return;
    int row = i >> 5;
    int c0  = (i & 31) << 2;
    float s = din[row];
    float4 v = h4[i];
    v.x = v.x * s + bg[c0 + 0];
    v.y = v.y * s + bg[c0 + 1];
    v.z = v.z * s + bg[c0 + 2];
    v.w = v.w * s + bg[c0 + 3];
    h4[i] = v;
}

// ---------------------------------------------------------------- dense GEMM
// out[N x NCOL] = act(A[N x 128] @ W[128 x NCOL] + bias), 16 rows/block,
// one wave per 16-col tile, fp32 WMMA 16x16x4 over K=128.
template <int NCOL, bool RELU>
__global__ __launch_bounds__(32 * (NCOL / 16))
void gemm_dense_kernel(const float* __restrict__ A, const float* __restrict__ W,
                       const float* __restrict__ bias, float* __restrict__ out, int N) {
    constexpr int NW = NCOL / 16;
    constexpr int T  = 32 * NW;
    __shared__ float At[16][ASTR];

    const int tid  = threadIdx.x;
    const int row0 = blockIdx.x * 16;

    // stage A tile (16 x 128 f32) into LDS, coalesced float4
    for (int i = tid; i < 16 * FEAT / 4; i += T) {
        int r  = i / (FEAT / 4);
        int c4 = i % (FEAT / 4);
        float4 v = make_float4(0.f, 0.f, 0.f, 0.f);
        if (row0 + r < N)
            v = ((const float4*)(A + (size_t)(row0 + r) * FEAT))[c4];
        At[r][c4 * 4 + 0] = v.x;
        At[r][c4 * 4 + 1] = v.y;
        At[r][c4 * 4 + 2] = v.z;
        At[r][c4 * 4 + 3] = v.w;
    }
    __syncthreads();

    const int lane = tid & 31;
    const int wave = tid >> 5;
    const int n0   = wave * 16;
    const int hl   = lane >> 4;     // which half of the wave
    const int mn   = lane & 15;     // row of A / col of B

    v8f acc = {0.f, 0.f, 0.f, 0.f, 0.f, 0.f, 0.f, 0.f};
#pragma unroll
    for (int k = 0; k < FEAT; k += 4) {
        const int kk = k + 2 * hl;
        const float2 af = *reinterpret_cast<const float2*>(&At[mn][kk]);
        v2f a = {af.x, af.y};
        v2f b = {W[(size_t)kk * NCOL + n0 + mn],
                 W[(size_t)(kk + 1) * NCOL + n0 + mn]};
        acc = __builtin_amdgcn_wmma_f32_16x16x4_f32(false, a, false, b,
                                                    (short)0, acc, false, false);
    }

    const float bb = bias[n0 + mn];
#pragma unroll
    for (int v = 0; v < 8; ++v) {
        int r = v + 8 * hl;
        if (row0 + r < N) {
            float val = acc[v] + bb;
            if (RELU) val = fmaxf(val, 0.f);
            out[(size_t)(row0 + r) * NCOL + n0 + mn] = val;
        }
    }
}

// ---------------------------------------------------------------- GEMM + MaxK
// t = A @ W + bias; keep exact top-32 per row (ties -> lower index, matching
// jax.lax.top_k); emit compact (val * deg_out^-1/2, col) pairs.
__global__ __launch_bounds__(256)
void gemm_maxk_kernel(const float* __restrict__ A, const float* __restrict__ W,
                      const float* __restrict__ bias, const float* __restrict__ dou,
                      float* __restrict__ cval, int* __restrict__ cidx, int N) {
    __shared__ float At[16][ASTR];

    const int tid  = threadIdx.x;
    const int row0 = blockIdx.x * 16;

    for (int i = tid; i < 16 * FEAT / 4; i += 256) {
        int r  = i / (FEAT / 4);
        int c4 = i % (FEAT / 4);
        float4 v = make_float4(0.f, 0.f, 0.f, 0.f);
        if (row0 + r < N)
            v = ((const float4*)(A + (size_t)(row0 + r) * FEAT))[c4];
        At[r][c4 * 4 + 0] = v.x;
        At[r][c4 * 4 + 1] = v.y;
        At[r][c4 * 4 + 2] = v.z;
        At[r][c4 * 4 + 3] = v.w;
    }
    __syncthreads();

    const int lane = tid & 31;
    const int wave = tid >> 5;
    const int n0   = wave * 16;
    const int hl   = lane >> 4;
    const int mn   = lane & 15;

    v8f acc = {0.f, 0.f, 0.f, 0.f, 0.f, 0.f, 0.f, 0.f};
#pragma unroll
    for (int k = 0; k < FEAT; k += 4) {
        const int kk = k + 2 * hl;
        const float2 af = *reinterpret_cast<const float2*>(&At[mn][kk]);
        v2f a = {af.x, af.y};
        v2f b = {W[(size_t)kk * FEAT + n0 + mn],
                 W[(size_t)(kk + 1) * FEAT + n0 + mn]};
        acc = __builtin_amdgcn_wmma_f32_16x16x4_f32(false, a, false, b,
                                                    (short)0, acc, false, false);
    }

    // write the 16x128 block (with bias) back into LDS for row-wise top-k
    __syncthreads();   // all waves done reading At
    const float bb = bias[n0 + mn];
#pragma unroll
    for (int v = 0; v < 8; ++v)
        At[v + 8 * hl][n0 + mn] = acc[v] + bb;
    __syncthreads();

    // each wave handles 2 rows; lane owns cols {lane, lane+32, lane+64, lane+96}
    for (int rr = 0; rr < 2; ++rr) {
        const int r = wave * 2 + rr;
        if (row0 + r >= N) continue;

        float vals[4];
#pragma unroll
        for (int j = 0; j < 4; ++j) vals[j] = At[r][lane + 32 * j];

        unsigned keep = 0u, taken = 0u;
        for (int it = 0; it < KSEL; ++it) {
            float bv = -INFINITY;
            int   bi = 1 << 30;
#pragma unroll
            for (int j = 0; j < 4; ++j) {
                if (!((taken >> j) & 1u)) {
                    int col = lane + 32 * j;
                    if (vals[j] > bv || (vals[j] == bv && col < bi)) { bv = vals[j]; bi = col; }
                }
            }
#pragma unroll
            for (int off = 16; off > 0; off >>= 1) {
                float ov = __shfl_xor(bv, off, 32);
                int   oi = __shfl_xor(bi, off, 32);
                if (ov > bv || (ov == bv && oi < bi)) { bv = ov; bi = oi; }
            }
            if ((bi & 31) == lane) {
                int j = bi >> 5;
                taken |= 1u << j;
                keep  |= 1u << j;
            }
        }

        // ballot-compacted write of exactly 32 (val, idx) pairs, scaled by deg_out^-1/2
        const float scale = dou[row0 + r];
        const size_t base_off = (size_t)(row0 + r) * KSEL;
        int base = 0;
#pragma unroll
        for (int j = 0; j < 4; ++j) {
            unsigned m = (unsigned)__ballot((keep >> j) & 1u);
            int below  = __popc(m & ((1u << lane) - 1u));
            if ((keep >> j) & 1u) {
                int o = base + below;
                cval[base_off + o] = vals[j] * scale;
                cidx[base_off + o] = lane + 32 * j;
            }
            base += __popc(m);
        }
    }
}

// ---------------------------------------------------------------- sparse SpMM
// one wave32 per edge: 32 compact nonzeros scatter-added into agg[dst] row
__global__ __launch_bounds__(256)
void spmm_scatter_kernel(const float* __restrict__ cval, const int* __restrict__ cidx,
                         const int* __restrict__ src, const int* __restrict__ dst,
                         float* __restrict__ agg, int E) {
    int w    = (blockIdx.x * blockDim.x + threadIdx.x) >> 5;
    int lane = threadIdx.x & 31;
    if (w >= E) return;
    int s = src[w];
    int d = dst[w];
    __builtin_prefetch(&agg[(size_t)d * FEAT], 1, 3);   // global_prefetch_b8
    float v = cval[(size_t)s * KSEL + lane];
    int   c = cidx[(size_t)s * KSEL + lane];
    atomicAdd(&agg[(size_t)d * FEAT + c], v);
}

// ---------------------------------------------------------------- launcher
extern "C" void kernel_launch(void* const* d_in, const int* in_sizes, int n_in,
                              void* d_out, int out_size, void* d_ws, size_t ws_size,
                              hipStream_t stream) {
    const float* x    = (const float*)d_in[0];
    const int*   src  = (const int*)d_in[1];
    const int*   dst  = (const int*)d_in[2];
    const float* W_in = (const float*)d_in[3];
    const float* b_in = (const float*)d_in[4];
    const float* W1   = (const float*)d_in[5];
    const float* b1   = (const float*)d_in[6];
    const float* bg1  = (const float*)d_in[7];
    const float* W2   = (const float*)d_in[8];
    const float* b2   = (const float*)d_in[9];
    const float* bg2  = (const float*)d_in[10];
    const float* Wo   = (const float*)d_in[11];
    const float* bo   = (const float*)d_in[12];

    const int N = in_sizes[0] / FEAT;
    const int E = in_sizes[1];

    // workspace layout
    float* h    = (float*)d_ws;                         // N*128 f32
    float* cval = h + (size_t)N * FEAT;                 // N*32  f32
    int*   cidx = (int*)(cval + (size_t)N * KSEL);      // N*32  i32
    float* dou  = (float*)(cidx + (size_t)N * KSEL);    // N     f32 (out-deg^-1/2)
    float* din  = dou + N;                              // N     f32 (in-deg^-1/2)

    const int gemm_blocks = (N + 15) / 16;
    const int n4          = N * FEAT / 4;               // float4 count of h
    const int z4_blocks   = (n4 + 255) / 256;
    const int edge_blocks = (E + 255) / 256;
    const int node_blocks = (N + 255) / 256;
    const int spmm_blocks = (int)(((long long)E * 32 + 255) / 256);

    // degrees
    zero1_kernel<<<(2 * N + 255) / 256, 256, 0, stream>>>(dou, 2 * N);
    degree_count_kernel<<<edge_blocks, 256, 0, stream>>>(src, dst, dou, din, E);
    degree_finalize_kernel<<<node_blocks, 256, 0, stream>>>(dou, din, N);

    // h = relu(x @ W_in + b_in)
    gemm_dense_kernel<FEAT, true><<<gemm_blocks, 256, 0, stream>>>(x, W_in, b_in, h, N);

    // layer 1
    gemm_maxk_kernel<<<gemm_blocks, 256, 0, stream>>>(h, W1, b1, dou, cval, cidx, N);
    zero4_kernel<<<z4_blocks, 256, 0, stream>>>((float4*)h, n4);
    spmm_scatter_kernel<<<spmm_blocks, 256, 0, stream>>>(cval, cidx, src, dst, h, E);
    spmm_finalize_kernel<<<z4_blocks, 256, 0, stream>>>((float4*)h, din, bg1, n4);

    // layer 2
    gemm_maxk_kernel<<<gemm_blocks, 256, 0, stream>>>(h, W2, b2, dou, cval, cidx, N);
    zero4_kernel<<<z4_blocks, 256, 0, stream>>>((float4*)h, n4);
    spmm_scatter_kernel<<<spmm_blocks, 256, 0, stream>>>(cval, cidx, src, dst, h, E);
    spmm_finalize_kernel<<<z4_blocks, 256, 0, stream>>>((float4*)h, din, bg2, n4);

    // out = h @ W_out + b_out
    gemm_dense_kernel<OUTF, false><<<gemm_blocks, 128, 0, stream>>>(h, Wo, bo, (float*)d_out, N);
}